// OptimizedParallelECASimAM_35235911696627
// MI455X (gfx1250) — compile-verified
//
#include <hip/hip_runtime.h>
#include <math.h>

#define BB   32
#define CC   256
#define HH   64
#define WW   64
#define HW   (HH * WW)     // 4096
#define NBC  (BB * CC)     // 8192
#define HID  64
#define E_LAMBDA 1e-4f

typedef float v2f __attribute__((ext_vector_type(2)));
typedef float v8f __attribute__((ext_vector_type(8)));

__device__ __forceinline__ float sigmoidf(float v) {
    return 1.0f / (1.0f + __expf(-v));
}

// Block-wide sum over 256 threads (8 waves of 32), result broadcast to all.
__device__ __forceinline__ float block_reduce_sum(float v, float* red) {
    #pragma unroll
    for (int off = 16; off >= 1; off >>= 1)
        v += __shfl_xor(v, off, 32);
    const int lane = threadIdx.x & 31;
    const int wv   = threadIdx.x >> 5;
    if (lane == 0) red[wv] = v;
    __syncthreads();
    if (wv == 0) {
        float s = (lane < 8) ? red[lane] : 0.0f;
        #pragma unroll
        for (int off = 4; off >= 1; off >>= 1)
            s += __shfl_xor(s, off, 32);
        if (lane == 0) red[0] = s;
    }
    __syncthreads();
    float r = red[0];
    __syncthreads();   // allow red[] reuse by a following reduction
    return r;
}

// ---------------------------------------------------------------------------
// Kernel 1: per-(b,c) mean, 1/(4(var+lambda)), and mean(x * sigmoid(energy)).
// One block per (b,c); each thread holds its 16 elements in registers so the
// three reductions cost a single read of x.
// ---------------------------------------------------------------------------
__global__ void __launch_bounds__(256)
stats_kernel(const float* __restrict__ x,
             float* __restrict__ mean_out,
             float* __restrict__ inv4_out,
             float* __restrict__ simam_out) {
    __shared__ float red[8];
    const int bc = blockIdx.x;
    const int t  = threadIdx.x;
    const float4* xp = (const float4*)(x + (size_t)bc * HW);

    float4 v[4];
    float s = 0.0f;
    #pragma unroll
    for (int i = 0; i < 4; ++i) {
        v[i] = xp[t + i * 256];
        s += v[i].x + v[i].y + v[i].z + v[i].w;
    }
    s = block_reduce_sum(s, red);
    const float m = s * (1.0f / HW);

    float ss = 0.0f;
    #pragma unroll
    for (int i = 0; i < 4; ++i) {
        float dx = v[i].x - m, dy = v[i].y - m, dz = v[i].z - m, dw = v[i].w - m;
        ss += dx * dx + dy * dy + dz * dz + dw * dw;
    }
    ss = block_reduce_sum(ss, red);
    const float var  = ss * (1.0f / (HW - 1));
    const float inv4 = 1.0f / (4.0f * (var + E_LAMBDA));

    float sm = 0.0f;
    #pragma unroll
    for (int i = 0; i < 4; ++i) {
        float e;
        e = v[i].x - m; sm += v[i].x * sigmoidf(e * e * inv4 + 0.5f);
        e = v[i].y - m; sm += v[i].y * sigmoidf(e * e * inv4 + 0.5f);
        e = v[i].z - m; sm += v[i].z * sigmoidf(e * e * inv4 + 0.5f);
        e = v[i].w - m; sm += v[i].w * sigmoidf(e * e * inv4 + 0.5f);
    }
    sm = block_reduce_sum(sm, red);

    if (t == 0) {
        mean_out[bc]  = m;
        inv4_out[bc]  = inv4;
        simam_out[bc] = sm * (1.0f / HW);
    }
}

// ---------------------------------------------------------------------------
// Kernel 2: ECA gate (K=5 channel conv + sigmoid), fusion GAP, and the
// 2-layer MLP via V_WMMA_F32_16X16X4_F32. Single 256-thread workgroup.
//   GEMM1: hid[32,64]   = relu(gap[32,256] @ w1[64,256]^T)
//   GEMM2: alpha[32,256]= sigmoid(hid @ w2[256,64]^T)
// WMMA f32 layouts (wave32):
//   A 16x4 : lane L -> row M=L%16, K=(L/16)*2 + {0,1} in v2f
//   B 4x16 : lane L -> col N=L%16, K=(L/16)*2 + {0,1} in v2f
//   D 16x16: lane L -> col N=L%16, row M=(L/16)*8 + i for acc[i]
// ---------------------------------------------------------------------------
#define GAP_LD (CC + 1)    // pad LDS rows to avoid 64-bank conflicts
#define HID_LD (HID + 1)

__global__ void __launch_bounds__(256)
gate_mlp_kernel(const float* __restrict__ mean,
                const float* __restrict__ simam_mean,
                const float* __restrict__ convw,
                const float* __restrict__ w1,
                const float* __restrict__ w2,
                float* __restrict__ eca_gate,
                float* __restrict__ alpha_out) {
    __shared__ float gap[BB * GAP_LD];   // ~32.1 KB
    __shared__ float hid[BB * HID_LD];   // ~8.1 KB
    const int t = threadIdx.x;

    const float k0 = convw[0], k1 = convw[1], k2 = convw[2],
                k3 = convw[3], k4 = convw[4];

    // ECA conv over channel dim of per-channel means, sigmoid gate, fusion GAP.
    for (int i = t; i < NBC; i += 256) {
        const int b = i >> 8;
        const int c = i & 255;
        const int base = b * CC;
        float acc = k2 * mean[base + c];
        if (c >= 2)      acc += k0 * mean[base + c - 2];
        if (c >= 1)      acc += k1 * mean[base + c - 1];
        if (c + 1 < CC)  acc += k3 * mean[base + c + 1];
        if (c + 2 < CC)  acc += k4 * mean[base + c + 2];
        const float g = sigmoidf(acc);
        eca_gate[i] = g;
        // gap = mean(x_eca + x_simam) = g*mean + mean(x*sigmoid(energy))
        gap[b * GAP_LD + c] = g * mean[i] + simam_mean[i];
    }
    __syncthreads();

    const int lane = t & 31;
    const int wv   = t >> 5;        // 0..7
    const int N    = lane & 15;
    const int half = lane >> 4;     // 0 or 1

    // ---- GEMM1: 2x4 tiles of 16x16, one per wave; K=256 in steps of 4 ----
    {
        const int m0 = (wv >> 2) * 16;
        const int n0 = (wv & 3) * 16;
        v8f acc = {};
        for (int kk = 0; kk < CC; kk += 4) {
            const int kb = kk + half * 2;
            v2f a, b;
            a.x = gap[(m0 + N) * GAP_LD + kb];
            a.y = gap[(m0 + N) * GAP_LD + kb + 1];
            // B[k][n] = w1[(n0+n)*CC + k]  (w1 is [HID, CC] row-major)
            b.x = w1[(n0 + N) * CC + kb];
            b.y = w1[(n0 + N) * CC + kb + 1];
            acc = __builtin_amdgcn_wmma_f32_16x16x4_f32(
                false, a, false, b, (short)0, acc, false, false);
        }
        #pragma unroll
        for (int i = 0; i < 8; ++i) {
            const float v = acc[i];
            hid[(m0 + half * 8 + i) * HID_LD + (n0 + N)] = v > 0.0f ? v : 0.0f;
        }
    }
    __syncthreads();

    // ---- GEMM2: 2x16 tiles of 16x16; each wave does 4; K=64 in steps of 4 ----
    for (int rep = 0; rep < 4; ++rep) {
        const int idx = rep * 8 + wv;      // 0..31
        const int m0  = (idx >> 4) * 16;
        const int n0  = (idx & 15) * 16;
        v8f acc = {};
        for (int kk = 0; kk < HID; kk += 4) {
            const int kb = kk + half * 2;
            v2f a, b;
            a.x = hid[(m0 + N) * HID_LD + kb];
            a.y = hid[(m0 + N) * HID_LD + kb + 1];
            // B[k][n] = w2[(n0+n)*HID + k]  (w2 is [CC, HID] row-major)
            b.x = w2[(n0 + N) * HID + kb];
            b.y = w2[(n0 + N) * HID + kb + 1];
            acc = __builtin_amdgcn_wmma_f32_16x16x4_f32(
                false, a, false, b, (short)0, acc, false, false);
        }
        #pragma unroll
        for (int i = 0; i < 8; ++i) {
            const int row = m0 + half * 8 + i;   // b index
            const int col = n0 + N;              // c index
            alpha_out[row * CC + col] = sigmoidf(acc[i]);
        }
    }
}

// ---------------------------------------------------------------------------
// Kernel 3: out = x * (alpha * sigmoid(energy) + (1-alpha) * eca_gate)
// One block per (b,c); per-block scalars become s_loads; float4 traffic.
// ---------------------------------------------------------------------------
__global__ void __launch_bounds__(256)
fuse_kernel(const float* __restrict__ x,
            const float* __restrict__ mean,
            const float* __restrict__ inv4,
            const float* __restrict__ eca_gate,
            const float* __restrict__ alpha,
            float* __restrict__ out) {
    const int bc = blockIdx.x;
    const float m  = mean[bc];
    const float i4 = inv4[bc];
    const float g  = eca_gate[bc];
    const float a  = alpha[bc];
    const float ag = (1.0f - a) * g;

    const float4* xp = (const float4*)(x   + (size_t)bc * HW);
    float4*       op = (float4*)      (out + (size_t)bc * HW);
    const int t = threadIdx.x;
    #pragma unroll
    for (int i = 0; i < 4; ++i) {
        float4 v = xp[t + i * 256];
        float4 r;
        float e;
        e = v.x - m; r.x = v.x * (a * sigmoidf(e * e * i4 + 0.5f) + ag);
        e = v.y - m; r.y = v.y * (a * sigmoidf(e * e * i4 + 0.5f) + ag);
        e = v.z - m; r.z = v.z * (a * sigmoidf(e * e * i4 + 0.5f) + ag);
        e = v.w - m; r.w = v.w * (a * sigmoidf(e * e * i4 + 0.5f) + ag);
        op[t + i * 256] = r;
    }
}

extern "C" void kernel_launch(void* const* d_in, const int* in_sizes, int n_in,
                              void* d_out, int out_size, void* d_ws, size_t ws_size,
                              hipStream_t stream) {
    const float* x     = (const float*)d_in[0];
    const float* convw = (const float*)d_in[1];
    const float* w1    = (const float*)d_in[2];
    const float* w2    = (const float*)d_in[3];
    float* out = (float*)d_out;

    float* ws        = (float*)d_ws;
    float* mean_buf  = ws + 0 * NBC;
    float* inv4_buf  = ws + 1 * NBC;
    float* simam_buf = ws + 2 * NBC;
    float* eca_buf   = ws + 3 * NBC;
    float* alpha_buf = ws + 4 * NBC;

    stats_kernel<<<NBC, 256, 0, stream>>>(x, mean_buf, inv4_buf, simam_buf);
    gate_mlp_kernel<<<1, 256, 0, stream>>>(mean_buf, simam_buf, convw, w1, w2,
                                           eca_buf, alpha_buf);
    fuse_kernel<<<NBC, 256, 0, stream>>>(x, mean_buf, inv4_buf, eca_buf,
                                         alpha_buf, out);
}